// ManifoldMessagePassing_53953379172480
// MI455X (gfx1250) — compile-verified
//
#include <hip/hip_runtime.h>

// ---------------------------------------------------------------------------
// ManifoldMessagePassing for MI455X (gfx1250, wave32).
//
//   s[n]  = sum_{e: dst[e]==n} ||edge_attr[e]||_2 * x[src[e]]   (scatter, f32 atomics)
//   out   = x @ psi^T + s @ phi^T                               (WMMA f32 16x16x4)
//
// N_NODES = 50000 (= 3125 * 16, exact), IN=OUT=64, E = 800000.
// ---------------------------------------------------------------------------

typedef __attribute__((ext_vector_type(2))) float v2f;
typedef __attribute__((ext_vector_type(8))) float v8f;

#define N_NODES 50000
#define N_EDGES 800000
#define CH      64

// ---- Kernel 0: zero the scatter accumulator (ws is poisoned 0xAA) ----------
__global__ __launch_bounds__(256) void mmp_zero(float4* __restrict__ s, int n4) {
    int i = blockIdx.x * blockDim.x + threadIdx.x;
    if (i < n4) s[i] = make_float4(0.f, 0.f, 0.f, 0.f);
}

// ---- Kernel 1: edge scatter  s[dst] += norm(edge_attr) * x[src] ------------
// One wave32 per edge; lane handles channels 2*lane and 2*lane+1.
// src/dst loads are wave-uniform -> single L2 request; x rows are L2-resident.
__global__ __launch_bounds__(256) void mmp_scatter(const float* __restrict__ x,
                                                   const long long* __restrict__ ei,
                                                   const float* __restrict__ ea,
                                                   float* __restrict__ s) {
    int gid  = blockIdx.x * blockDim.x + threadIdx.x;
    int e    = gid >> 5;
    if (e >= N_EDGES) return;
    int lane = gid & 31;

    long long src = ei[e];
    long long dst = ei[N_EDGES + e];

    float a0 = ea[3 * e + 0];
    float a1 = ea[3 * e + 1];
    float a2 = ea[3 * e + 2];
    float nrm = sqrtf(a0 * a0 + a1 * a1 + a2 * a2);

    const v2f* xr = (const v2f*)(x + (size_t)src * CH + 2 * lane);
    float*     sr = s + (size_t)dst * CH + 2 * lane;
    v2f xv = *xr;
    atomicAdd(sr + 0, nrm * xv.x);
    atomicAdd(sr + 1, nrm * xv.y);
}

// ---- Kernel 2: out = x @ psi^T + s @ phi^T  via V_WMMA_F32_16X16X4_F32 -----
// One wave per 16-node tile, full 64 output channels (4 accumulator tiles).
// A frag: lane l -> row l%16, K = (l/16)*2 .. +1  (float2 load from src rows)
// B frag: lane l -> col l%16 (+16*nt), same K pair; B[k][o]=W[o][k] so the
//         load is a contiguous float2 from the row-major [64,64] weight.
// D:      lane l, vgpr r -> row r + 8*(l/16), col l%16.
__global__ __launch_bounds__(256) void mmp_gemm(const float* __restrict__ x,
                                                const float* __restrict__ s,
                                                const float* __restrict__ phi,
                                                const float* __restrict__ psi,
                                                float* __restrict__ out) {
    const int wave  = blockIdx.x * (blockDim.x >> 5) + (threadIdx.x >> 5);
    const int tile0 = wave * 16;
    if (tile0 >= N_NODES) return;           // uniform per wave: EXEC stays all-1s

    const int lane = threadIdx.x & 31;
    const int mrow = lane & 15;             // row within tile / col within N-tile
    const int kg   = lane >> 4;             // K-group: 0 or 1

    v8f acc[4] = {v8f{}, v8f{}, v8f{}, v8f{}};

    #pragma unroll
    for (int p = 0; p < 2; ++p) {
        const float* A = p ? s : x;         // pass 0: psi*x, pass 1: phi*s
        const float* W = p ? phi : psi;
        const float* arow  = A + (size_t)(tile0 + mrow) * CH + 2 * kg;
        const float* wbase = W + (size_t)mrow * CH + 2 * kg;
        #pragma unroll
        for (int k = 0; k < CH; k += 4) {
            v2f a = *(const v2f*)(arow + k);
            #pragma unroll
            for (int nt = 0; nt < 4; ++nt) {
                v2f b = *(const v2f*)(wbase + (size_t)nt * 16 * CH + k);
                acc[nt] = __builtin_amdgcn_wmma_f32_16x16x4_f32(
                    /*neg_a=*/false, a, /*neg_b=*/false, b,
                    /*c_mod=*/(short)0, acc[nt],
                    /*reuse_a=*/false, /*reuse_b=*/false);
            }
        }
    }

    #pragma unroll
    for (int nt = 0; nt < 4; ++nt) {
        #pragma unroll
        for (int r = 0; r < 8; ++r) {
            int orow = tile0 + r + 8 * kg;
            out[(size_t)orow * CH + nt * 16 + mrow] = acc[nt][r];
        }
    }
}

// ---------------------------------------------------------------------------
extern "C" void kernel_launch(void* const* d_in, const int* in_sizes, int n_in,
                              void* d_out, int out_size, void* d_ws, size_t ws_size,
                              hipStream_t stream) {
    const float*     x    = (const float*)d_in[0];
    const long long* ei   = (const long long*)d_in[1];   // int64 [2, E]
    const float*     ea   = (const float*)d_in[2];       // [E, 3]
    const float*     phiw = (const float*)d_in[3];       // [64, 64]
    const float*     psiw = (const float*)d_in[4];       // [64, 64]
    float*           out  = (float*)d_out;               // [N, 64]
    float*           s    = (float*)d_ws;                // [N, 64] scatter accum

    // 1) zero the accumulator (ws is poisoned before timing)
    {
        int n4 = N_NODES * CH / 4;
        mmp_zero<<<(n4 + 255) / 256, 256, 0, stream>>>((float4*)s, n4);
    }
    // 2) edge scatter: 32 threads per edge
    {
        long long threads = (long long)N_EDGES * 32;
        int blocks = (int)((threads + 255) / 256);
        mmp_scatter<<<blocks, 256, 0, stream>>>(x, ei, ea, s);
    }
    // 3) fused dual-GEMM via WMMA: 3125 waves, 8 waves per block
    {
        int n_waves = N_NODES / 16;                       // 3125
        int blocks  = (n_waves + 7) / 8;                  // 391
        mmp_gemm<<<blocks, 256, 0, stream>>>(x, s, phiw, psiw, out);
    }
}